// ResidualBlock_24386824306904
// MI455X (gfx1250) — compile-verified
//
#include <hip/hip_runtime.h>
#include <hip/hip_bf16.h>
#include <math.h>

// Problem constants (from the reference harness)
#define DIM   64            // node/edge feature dim
#define NGRAPH 512          // number of graphs
#define LN_EPS 1e-5f

typedef __attribute__((ext_vector_type(2))) float v2f;
typedef __attribute__((ext_vector_type(8))) float v8f;

// ---------------------------------------------------------------------------
// Kernel 0: zero workspace (agg accumulator + graph stats). Needed every call
// because the harness poisons ws once and never re-zeroes between replays.
// ---------------------------------------------------------------------------
__global__ __launch_bounds__(256) void zero_ws_kernel(float* p, int n) {
    int i = blockIdx.x * 256 + threadIdx.x;
    if (i < n) p[i] = 0.0f;
}

// ---------------------------------------------------------------------------
// Kernel 1: fused edge pipeline.
//   ea  = silu(edge_attr @ We + be)          (WMMA GEMM 1)
//   h   = relu(x[src] + ea)                  (gather, L2-resident x, RT hint)
//   msg = silu(h @ Wm + bm)                  (WMMA GEMM 2)
//   agg[dst] += msg                          (f32 atomics, L2-resident agg)
//
// Streamed-once data (edge_attr, edge_index) uses non-temporal loads so the
// 409 MB edge stream does not evict the hot x/agg lines from the 192 MB L2.
//
// 128 threads = 4 waves. Each wave processes TPW tiles of 16 edges.
// Weights are pre-swizzled into LDS so each WMMA B fragment is one ds_load_b64.
//
// f32 WMMA layouts (wave32), 16x16x4:
//   A (16x4):  lane&15 = row M, (lane>>4)*2 = K offset, .x/.y = K, K+1
//   B (4x16):  lane&15 = col N, (lane>>4)*2 = K offset, .x/.y = K, K+1
//   C/D:       vgpr r -> M = r + (lane>>4)*8, N = lane&15
// ---------------------------------------------------------------------------
#define WAVES 4
#define TPW   8          // 16-edge tiles per wave -> 512 edges per block
#define HSTR  65         // padded row stride for the transpose tile

__global__ __launch_bounds__(128) void edge_gemm_kernel(
    const float* __restrict__ x,
    const int*   __restrict__ src,
    const int*   __restrict__ dst,
    const float* __restrict__ edge_attr,
    const float* __restrict__ We, const float* __restrict__ be,
    const float* __restrict__ Wm, const float* __restrict__ bm,
    float* __restrict__ agg, int E_)
{
    __shared__ v2f   wea[2048];            // We in B-fragment order (16 KB)
    __shared__ v2f   wma[2048];            // Wm in B-fragment order (16 KB)
    __shared__ float blds[2 * DIM];        // be | bm
    __shared__ float hlds[WAVES * 16 * HSTR];  // per-wave transpose tiles

    const int tid = threadIdx.x;

    // Swizzle weights into LDS: slot s = (kc, nt, lane)
    for (int i = 0; i < 16; ++i) {
        int s    = tid + i * 128;
        int kc   = s >> 7;
        int nt   = (s >> 5) & 3;
        int ln   = s & 31;
        int koff = (ln >> 4) * 2;
        int brow = kc * 4 + koff;
        int col  = nt * 16 + (ln & 15);
        wea[s] = v2f{We[brow * DIM + col], We[(brow + 1) * DIM + col]};
        wma[s] = v2f{Wm[brow * DIM + col], Wm[(brow + 1) * DIM + col]};
    }
    if (tid < DIM)            blds[tid] = be[tid];
    else if (tid < 2 * DIM)   blds[tid] = bm[tid - DIM];
    __syncthreads();

    const int wid  = tid >> 5;
    const int lane = tid & 31;
    const int row  = lane & 15;          // M row / edge-in-tile this lane owns
    const int nlo  = lane & 15;          // N column for B/C/D
    const int koff = (lane >> 4) * 2;    // K sub-offset for A/B frags
    float* ht = &hlds[wid * 16 * HSTR];

    const int tile0 = (blockIdx.x * WAVES + wid) * TPW;

    for (int t = 0; t < TPW; ++t) {
        const int e0 = (tile0 + t) * 16;
        if (e0 >= E_) break;             // E is a multiple of 16 in this harness

        // Prefetch next tile's A stream (global_prefetch_b8) to cover HBM
        // latency of the back-to-back dependent WMMA chain.
        if (e0 + 16 < E_) {
            __builtin_prefetch(edge_attr + (size_t)(e0 + 16 + row) * DIM + koff, 0, 1);
        }

        // ---- A fragments: edge_attr tile (16 x 64), streamed (NT) -------
        v2f a[16];
        const float* ep = edge_attr + (size_t)(e0 + row) * DIM + koff;
        #pragma unroll
        for (int kc = 0; kc < 16; ++kc)
            a[kc] = __builtin_nontemporal_load((const v2f*)(ep + kc * 4));

        // ---- GEMM 1: ea = edge_attr @ We + be ---------------------------
        v8f acc[4];
        #pragma unroll
        for (int nt = 0; nt < 4; ++nt) {
            float b0 = blds[nt * 16 + nlo];
            acc[nt] = v8f{b0, b0, b0, b0, b0, b0, b0, b0};
        }
        #pragma unroll
        for (int nt = 0; nt < 4; ++nt) {
            #pragma unroll
            for (int kc = 0; kc < 16; ++kc) {
                acc[nt] = __builtin_amdgcn_wmma_f32_16x16x4_f32(
                    false, a[kc], false, wea[kc * 128 + nt * 32 + lane],
                    (short)0, acc[nt], false, false);
            }
        }

        // ---- SiLU, write C-layout -> padded LDS tile (transpose) --------
        #pragma unroll
        for (int nt = 0; nt < 4; ++nt) {
            #pragma unroll
            for (int r = 0; r < 8; ++r) {
                int m = r + (lane >> 4) * 8;
                float v = acc[nt][r];
                v = v / (1.0f + __expf(-v));
                ht[m * HSTR + nt * 16 + nlo] = v;
            }
        }

        // ---- h = relu(x[src] + ea) as A fragments -----------------------
        // edge_index is streamed once -> NT; x is hot in L2 -> default RT.
        const int sidx = __builtin_nontemporal_load(&src[e0 + row]);
        const float* xp = x + (size_t)sidx * DIM + koff;
        v2f h[16];
        #pragma unroll
        for (int kc = 0; kc < 16; ++kc) {
            v2f xv = *(const v2f*)(xp + kc * 4);
            float h0 = ht[row * HSTR + kc * 4 + koff]     + xv.x;
            float h1 = ht[row * HSTR + kc * 4 + koff + 1] + xv.y;
            h[kc] = v2f{fmaxf(h0, 0.0f), fmaxf(h1, 0.0f)};
        }

        // ---- GEMM 2: msg = h @ Wm + bm ----------------------------------
        v8f acc2[4];
        #pragma unroll
        for (int nt = 0; nt < 4; ++nt) {
            float b0 = blds[DIM + nt * 16 + nlo];
            acc2[nt] = v8f{b0, b0, b0, b0, b0, b0, b0, b0};
        }
        #pragma unroll
        for (int nt = 0; nt < 4; ++nt) {
            #pragma unroll
            for (int kc = 0; kc < 16; ++kc) {
                acc2[nt] = __builtin_amdgcn_wmma_f32_16x16x4_f32(
                    false, h[kc], false, wma[kc * 128 + nt * 32 + lane],
                    (short)0, acc2[nt], false, false);
            }
        }

        // ---- SiLU + scatter-add to agg[dst] (agg stays hot in L2) -------
        const int didx = __builtin_nontemporal_load(&dst[e0 + row]);
        #pragma unroll
        for (int nt = 0; nt < 4; ++nt) {
            #pragma unroll
            for (int r = 0; r < 8; ++r) {
                int m  = r + (lane >> 4) * 8;
                int dm = __shfl(didx, m, 32);
                float v = acc2[nt][r];
                v = v / (1.0f + __expf(-v));
                atomicAdd(&agg[(size_t)dm * DIM + nt * 16 + nlo], v);
            }
        }
    }
}

// ---------------------------------------------------------------------------
// Kernel 2: pre = agg + (1+eps)*x  (in place on agg), and per-graph
// sum / sumsq / count via wave-per-node reduction + 3 atomics.
// ---------------------------------------------------------------------------
__global__ __launch_bounds__(256) void node_stats_kernel(
    const float* __restrict__ x, const int* __restrict__ n2g,
    const float* __restrict__ eps_p,
    float* __restrict__ pre,        // in: agg, out: pre
    float* __restrict__ gsum, float* __restrict__ gsq, float* __restrict__ gcnt,
    int N_)
{
    const int wid = threadIdx.x >> 5, lane = threadIdx.x & 31;
    const int n = blockIdx.x * 8 + wid;
    if (n >= N_) return;
    const float k = 1.0f + eps_p[0];
    const size_t base = (size_t)n * DIM;
    float v0 = pre[base + lane]      + k * x[base + lane];
    float v1 = pre[base + lane + 32] + k * x[base + lane + 32];
    pre[base + lane]      = v0;
    pre[base + lane + 32] = v1;
    float s = v0 + v1;
    float q = v0 * v0 + v1 * v1;
    #pragma unroll
    for (int o = 16; o > 0; o >>= 1) {
        s += __shfl_xor(s, o, 32);
        q += __shfl_xor(q, o, 32);
    }
    if (lane == 0) {
        int g = n2g[n];
        atomicAdd(&gsum[g], s);
        atomicAdd(&gsq[g], q);
        atomicAdd(&gcnt[g], (float)DIM);   // count already includes *f
    }
}

// ---------------------------------------------------------------------------
// Kernel 3: per-graph mean / rstd.
// ---------------------------------------------------------------------------
__global__ __launch_bounds__(256) void finalize_stats_kernel(
    const float* gsum, const float* gsq, const float* gcnt,
    float* mean, float* rstd, int G_)
{
    int g = blockIdx.x * 256 + threadIdx.x;
    if (g >= G_) return;
    float c = fmaxf(gcnt[g], 1.0f);
    float m = gsum[g] / c;
    float v = gsq[g] / c - m * m;
    mean[g] = m;
    rstd[g] = rsqrtf(v + LN_EPS);
}

// ---------------------------------------------------------------------------
// Kernel 4: out = relu((pre - mean)*rstd*gamma + beta + x)
// pre is last-use here (NT load); out is write-once (NT store).
// ---------------------------------------------------------------------------
__global__ __launch_bounds__(256) void out_kernel(
    const float* __restrict__ pre, const float* __restrict__ x,
    const int* __restrict__ n2g,
    const float* __restrict__ mean, const float* __restrict__ rstd,
    const float* __restrict__ gamma, const float* __restrict__ beta,
    float* __restrict__ out, int total)
{
    int i = blockIdx.x * 256 + threadIdx.x;
    if (i >= total) return;
    int n = i >> 6, c = i & (DIM - 1);
    int g = n2g[n];
    float p = __builtin_nontemporal_load(&pre[i]);
    float v = (p - mean[g]) * rstd[g] * gamma[c] + beta[c] + x[i];
    __builtin_nontemporal_store(fmaxf(v, 0.0f), &out[i]);
}

// ---------------------------------------------------------------------------
// kernel_launch: branch 1 of the reference is dead (overwritten) -> skipped.
// ---------------------------------------------------------------------------
extern "C" void kernel_launch(void* const* d_in, const int* in_sizes, int n_in,
                              void* d_out, int out_size, void* d_ws, size_t ws_size,
                              hipStream_t stream) {
    const float* x     = (const float*)d_in[0];
    const int*   ei    = (const int*)  d_in[1];
    const float* eattr = (const float*)d_in[2];
    const int*   n2g   = (const int*)  d_in[3];
    // branch-2 params only (indices 11..17)
    const float* We2   = (const float*)d_in[11];
    const float* be2   = (const float*)d_in[12];
    const float* Wm2   = (const float*)d_in[13];
    const float* bm2   = (const float*)d_in[14];
    const float* eps2  = (const float*)d_in[15];
    const float* g2    = (const float*)d_in[16];
    const float* b2    = (const float*)d_in[17];

    const int N_ = in_sizes[0] / DIM;
    const int E_ = in_sizes[1] / 2;
    const int G_ = NGRAPH;
    const int* src = ei;
    const int* dst = ei + E_;

    // workspace layout
    float* agg  = (float*)d_ws;            // N*64 (becomes `pre` in place)
    float* gsum = agg + (size_t)N_ * DIM;  // G
    float* gsq  = gsum + G_;               // G
    float* gcnt = gsq + G_;                // G
    float* mean = gcnt + G_;               // G
    float* rstd = mean + G_;               // G

    // 0) zero agg + stats (mean/rstd are overwritten, no zero needed)
    {
        int zn = N_ * DIM + 3 * G_;
        zero_ws_kernel<<<(zn + 255) / 256, 256, 0, stream>>>(agg, zn);
    }

    // 1) fused edge GEMMs + scatter
    {
        int tiles  = (E_ + 15) / 16;
        int perBlk = WAVES * TPW;
        int blocks = (tiles + perBlk - 1) / perBlk;
        edge_gemm_kernel<<<blocks, 128, 0, stream>>>(
            x, src, dst, eattr, We2, be2, Wm2, bm2, agg, E_);
    }

    // 2) pre + per-graph stats
    node_stats_kernel<<<(N_ + 7) / 8, 256, 0, stream>>>(
        x, n2g, eps2, agg, gsum, gsq, gcnt, N_);

    // 3) mean / rstd
    finalize_stats_kernel<<<(G_ + 255) / 256, 256, 0, stream>>>(
        gsum, gsq, gcnt, mean, rstd, G_);

    // 4) normalize + residual + relu
    {
        int total = N_ * DIM;
        out_kernel<<<(total + 255) / 256, 256, 0, stream>>>(
            agg, x, n2g, mean, rstd, g2, b2, (float*)d_out, total);
    }
}